// TreecrfLoss_44659069943945
// MI455X (gfx1250) — compile-verified
//
#include <hip/hip_runtime.h>

// TreeCRF loss for MI455X (gfx1250, wave32, WMMA).
// CYK inside algorithm done as blocked log-semiring matmul:
//   far split points  -> scaled-exp f32 tile GEMMs via V_WMMA_F32_16X16X4_F32
//   near split points -> in-tile sequential wavefront in log space
// 128 independent charts (64 batch x {s_all, s_ob}), one workgroup each.

#define LEN 128
#define BSZ 64
#define NEGV -1000000000.0f
#define CS 132  // LDS chart row stride (132 mod 64 == 4 -> conflict-free strided reads)

typedef __attribute__((ext_vector_type(2))) float v2f;
typedef __attribute__((ext_vector_type(8))) float v8f;

__device__ __forceinline__ float logaddexpf_(float a, float b) {
  float m = fmaxf(a, b);
  float d = fabsf(a - b);
  return m + log1pf(__expf(-d));
}

// s_all (type==0) or s_ob (type==1) score at (i,j), computed on the fly from globals.
__device__ __forceinline__ float s_val(const float* __restrict__ sl,
                                       const int* __restrict__ si,
                                       int b, int type, int i, int j) {
  size_t base = ((size_t)b * LEN + i) * LEN + j;
  float l0 = sl[base * 2 + 0];
  float l1 = sl[base * 2 + 1];
  if (type == 0) return logaddexpf_(l0, l1);
  // gold = (spans_ind == 2) on the valid (upper) triangle
  bool gold = (si[base] == 2);
  float c0 = gold ? NEGV : l0;
  float c1 = gold ? l1 : NEGV;
  return logaddexpf_(c0, c1);
}

__global__ __launch_bounds__(256) void cyk_kernel(const float* __restrict__ sl,
                                                  const int* __restrict__ si,
                                                  float* __restrict__ ws) {
  __shared__ float chart[LEN * CS];  // log-space inside chart (upper triangle used)
  __shared__ float tileMax[64];      // per-16x16-tile max of finalized chart entries

  const int b    = blockIdx.x >> 1;
  const int type = blockIdx.x & 1;  // 0 = s_all (logZ, +), 1 = s_ob (marginals, -)
  const int tid  = threadIdx.x;
  const int wave = tid >> 5;
  const int lane = tid & 31;
  const int half = lane >> 4;
  const int ln   = lane & 15;

  // ---------------- band 0: diagonal tiles, one per wave ----------------
  {
    const int base = wave * 16;
    if (lane < 16) {
      int i = base + lane;
      chart[i * CS + i] = s_val(sl, si, b, type, i, i);
    }
    asm volatile("s_wait_dscnt 0" ::: "memory");
    for (int wd = 1; wd < 16; ++wd) {
      if (lane <= 15 - wd) {
        int i = base + lane;
        int j = i + wd;
        float mx = -3.4e38f;
        for (int k = i; k < j; ++k)
          mx = fmaxf(mx, chart[i * CS + k] + chart[(k + 1) * CS + j]);
        float sm = 0.f;
        for (int k = i; k < j; ++k)
          sm += __expf(chart[i * CS + k] + chart[(k + 1) * CS + j] - mx);
        chart[i * CS + j] = s_val(sl, si, b, type, i, j) + mx + __logf(sm);
      }
      asm volatile("s_wait_dscnt 0" ::: "memory");
    }
    // tile max over the valid (upper) half of the diagonal tile
    float m = -3.4e38f;
    int col = ln, r0 = half * 8;
    for (int r = r0; r < r0 + 8; ++r)
      if (r <= col) m = fmaxf(m, chart[(base + r) * CS + base + col]);
    for (int off = 16; off >= 1; off >>= 1)
      m = fmaxf(m, __shfl_xor(m, off, 32));
    if (lane == 0) tileMax[wave * 8 + wave] = m;
  }
  __syncthreads();

  // ---------------- bands d = 1..7: one dest tile per wave ----------------
  for (int d = 1; d < 8; ++d) {
    if (wave < 8 - d) {
      const int bi = wave, bj = wave + d;

      // Scales per far k-block. B tile rows straddle chart tiles (bk,bj),(bk+1,bj);
      // max of both tile maxima is a safe upper bound.
      float mBv[6];
      float S = -3.4e38f;
      for (int bk = bi + 1; bk < bj; ++bk) {
        float ma = tileMax[bi * 8 + bk];
        float mb = fmaxf(tileMax[bk * 8 + bj], tileMax[(bk + 1) * 8 + bj]);
        mBv[bk - bi - 1] = mb;
        S = fmaxf(S, ma + mb);
      }

      if (d >= 2) {
        // Far splits: accumulate exp-scaled tile products with f32 WMMA.
        v8f c = {0.f, 0.f, 0.f, 0.f, 0.f, 0.f, 0.f, 0.f};
        const int ia = bi * 16 + ln;  // A-fragment row (M = lane&15)
        const int jb = bj * 16 + ln;  // B-fragment col (N = lane&15)
        for (int bk = bi + 1; bk < bj; ++bk) {
          float mb = mBv[bk - bi - 1];
          float sA = S - mb;  // fold cross-block rescale into A's exponent
          for (int kc = 0; kc < 4; ++kc) {
            // 16x16x4 f32 layout: lanes 0-15 hold K = kc*4+{0,1}, lanes 16-31 K = kc*4+{2,3}
            int kg = bk * 16 + kc * 4 + 2 * half;
            v2f a, bb;
            a.x  = __expf(chart[ia * CS + kg]       - sA);
            a.y  = __expf(chart[ia * CS + kg + 1]   - sA);
            bb.x = __expf(chart[(kg + 1) * CS + jb] - mb);  // right operand = chart shifted by 1 row
            bb.y = __expf(chart[(kg + 2) * CS + jb] - mb);
            c = __builtin_amdgcn_wmma_f32_16x16x4_f32(false, a, false, bb,
                                                      (short)0, c, false, false);
          }
        }
        // Stash partial (log space) into the dest tile slots.
        for (int e = 0; e < 8; ++e) {
          int r  = bi * 16 + e + 8 * half;  // C/D layout: vgpr e -> M = e (+8 upper half)
          int cc = bj * 16 + ln;
          float v = c[e];
          chart[r * CS + cc] = (v > 0.f) ? (S + __logf(v)) : NEGV;
        }
      } else {
        for (int e = 0; e < 8; ++e) {
          int r  = bi * 16 + e + 8 * half;
          int cc = bj * 16 + ln;
          chart[r * CS + cc] = NEGV;
        }
      }
      asm volatile("s_wait_dscnt 0" ::: "memory");

      // Near splits: k in [i, 16bi+15] U [16bj, j-1]. Anti-diagonal wavefront order:
      // entry (ri,cj) at t = (15-ri)+cj depends only on already-finalized entries.
      for (int t = 0; t < 31; ++t) {
        int cj = lane;
        int ri = 15 - (t - cj);
        if (lane < 16 && ri >= 0 && ri <= 15) {
          int i = bi * 16 + ri;
          int j = bj * 16 + cj;
          float accLog = chart[i * CS + j];  // partial from far GEMMs (or NEGV)
          int kA1 = bi * 16 + 15;
          int kB0 = bj * 16;
          float mx = -3.4e38f;
          for (int k = i; k <= kA1; ++k)
            mx = fmaxf(mx, chart[i * CS + k] + chart[(k + 1) * CS + j]);
          for (int k = kB0; k < j; ++k)
            mx = fmaxf(mx, chart[i * CS + k] + chart[(k + 1) * CS + j]);
          float sm = 0.f;
          for (int k = i; k <= kA1; ++k)
            sm += __expf(chart[i * CS + k] + chart[(k + 1) * CS + j] - mx);
          for (int k = kB0; k < j; ++k)
            sm += __expf(chart[i * CS + k] + chart[(k + 1) * CS + j] - mx);
          float tot = logaddexpf_(accLog, mx + __logf(sm));
          chart[i * CS + j] = s_val(sl, si, b, type, i, j) + tot;
        }
        asm volatile("s_wait_dscnt 0" ::: "memory");
      }

      // tile max for later bands
      float m = -3.4e38f;
      int col = ln, r0 = half * 8;
      for (int r = r0; r < r0 + 8; ++r)
        m = fmaxf(m, chart[(bi * 16 + r) * CS + bj * 16 + col]);
      for (int off = 16; off >= 1; off >>= 1)
        m = fmaxf(m, __shfl_xor(m, off, 32));
      if (lane == 0) tileMax[bi * 8 + bj] = m;
    }
    __syncthreads();
  }

  if (tid == 0) ws[blockIdx.x] = chart[0 * CS + (LEN - 1)];  // root = chart[0][L-1]
}

// BCE-with-logits reductions for ph and pt; deterministic per-block partials.
__global__ __launch_bounds__(256) void bce_kernel(const float* __restrict__ ph,
                                                  const float* __restrict__ pt,
                                                  const int* __restrict__ phi,
                                                  const int* __restrict__ pti,
                                                  float* __restrict__ wsph,
                                                  float* __restrict__ wspt) {
  __shared__ float r1[256];
  __shared__ float r2[256];
  const size_t n = (size_t)BSZ * LEN * LEN;
  float a = 0.f, c = 0.f;
  for (size_t idx = (size_t)blockIdx.x * blockDim.x + threadIdx.x; idx < n;
       idx += (size_t)gridDim.x * blockDim.x) {
    float x = ph[idx];
    float y = (float)phi[idx];
    a += fmaxf(x, 0.f) - x * y + log1pf(__expf(-fabsf(x)));  // softplus(x) - x*y
    x = pt[idx];
    y = (float)pti[idx];
    c += fmaxf(x, 0.f) - x * y + log1pf(__expf(-fabsf(x)));
  }
  r1[threadIdx.x] = a;
  r2[threadIdx.x] = c;
  __syncthreads();
  for (int off = 128; off > 0; off >>= 1) {
    if (threadIdx.x < off) {
      r1[threadIdx.x] += r1[threadIdx.x + off];
      r2[threadIdx.x] += r2[threadIdx.x + off];
    }
    __syncthreads();
  }
  if (threadIdx.x == 0) {
    wsph[blockIdx.x] = r1[0];
    wspt[blockIdx.x] = r2[0];
  }
}

__global__ void finalize_kernel(const float* __restrict__ ws, float* __restrict__ out) {
  float spanSum = 0.f;
  for (int i = 0; i < 128; ++i) spanSum += (i & 1) ? -ws[i] : ws[i];  // +logZ - marginals
  float sph = 0.f, spt = 0.f;
  for (int i = 0; i < 256; ++i) {
    sph += ws[128 + i];
    spt += ws[384 + i];
  }
  const float denomS = (float)(BSZ * LEN);              // 8192
  const float nArc   = (float)BSZ * (float)(LEN * LEN); // 1048576
  out[0] = 0.5f * (spanSum / denomS) + 0.5f * (sph / nArc + spt / nArc);
}

extern "C" void kernel_launch(void* const* d_in, const int* in_sizes, int n_in,
                              void* d_out, int out_size, void* d_ws, size_t ws_size,
                              hipStream_t stream) {
  (void)in_sizes; (void)n_in; (void)out_size; (void)ws_size;
  const float* sl  = (const float*)d_in[0];  // span_logits (B,L,L,2) f32
  const float* ph  = (const float*)d_in[1];  // ph (B,L,L) f32
  const float* pt  = (const float*)d_in[2];  // pt (B,L,L) f32
  const int*   si  = (const int*)d_in[3];    // spans_ind (B,L,L) i32
  const int*   phi = (const int*)d_in[4];    // ph_ind i32
  const int*   pti = (const int*)d_in[5];    // pt_ind i32
  // d_in[6..8] are constant masks implied by the structure; not needed.
  float* ws  = (float*)d_ws;
  float* out = (float*)d_out;

  cyk_kernel<<<BSZ * 2, 256, 0, stream>>>(sl, si, ws);                 // ws[0..127]
  bce_kernel<<<256, 256, 0, stream>>>(ph, pt, phi, pti, ws + 128, ws + 384);
  finalize_kernel<<<1, 1, 0, stream>>>(ws, out);
}